// ProjectedAdaptiveLogSoftmax_11879879543667
// MI455X (gfx1250) — compile-verified
//
#include <hip/hip_runtime.h>
#include <hip/hip_bf16.h>
#include <math.h>

typedef __attribute__((ext_vector_type(2))) float v2f;
typedef __attribute__((ext_vector_type(8))) float v8f;
typedef int v4i_vs __attribute__((vector_size(16)));              // int4, as in builtin proto
typedef __attribute__((address_space(1))) v4i_vs as1_v4i;         // global
typedef __attribute__((address_space(3))) v4i_vs as3_v4i;         // LDS

#define N_ROWS 4096
#define D_EMB  1024
#define COLS_PER_BLOCK 4096
#define CHUNKS_PER_BLOCK (COLS_PER_BLOCK / 16)
#define NEG_BIG (-3.0e38f)

#if defined(__AMDGCN__) && __has_builtin(__builtin_amdgcn_global_load_async_to_lds_b128)
#define USE_ASYNC_LDS 1
#else
#define USE_ASYNC_LDS 0
#endif

__device__ __forceinline__ float redmax16(float v) {
#pragma unroll
  for (int m = 1; m < 16; m <<= 1) v = fmaxf(v, __shfl_xor(v, m, 32));
  return v;
}
__device__ __forceinline__ float redsum16(float v) {
#pragma unroll
  for (int m = 1; m < 16; m <<= 1) v += __shfl_xor(v, m, 32);
  return v;
}

// ---------------------------------------------------------------------------
// Pass 1: proj = H[4096 x K] @ P[K x Ncols], fp32 WMMA, one wave per 16x16 tile
// ---------------------------------------------------------------------------
__global__ void proj_gemm(const float* __restrict__ H, const float* __restrict__ P,
                          float* __restrict__ out, int K, int Ncols) {
  const int ntn  = Ncols >> 4;
  const int tile = blockIdx.x;
  const int rt   = tile / ntn;
  const int nt   = tile % ntn;
  const int lane = threadIdx.x & 31;
  const int M    = lane & 15;          // A row / B col / C col
  const int h    = lane >> 4;          // lane half selects K sub-pair

  const float* hrow = H + (size_t)(rt * 16 + M) * K + 2 * h;
  const float* pcol = P + (size_t)(2 * h) * Ncols + nt * 16 + M;

  v8f acc = {};
  for (int k0 = 0; k0 < K; k0 += 4) {
    v2f a = *(const v2f*)(hrow + k0);
    v2f b;
    b.x = pcol[(size_t)k0 * Ncols];
    b.y = pcol[(size_t)(k0 + 1) * Ncols];
    acc = __builtin_amdgcn_wmma_f32_16x16x4_f32(false, a, false, b, (short)0, acc,
                                                false, false);
  }
#pragma unroll
  for (int j = 0; j < 8; ++j)
    out[(size_t)(rt * 16 + j + 8 * h) * Ncols + nt * 16 + M] = acc[j];
}

// ---------------------------------------------------------------------------
// Pass 2: streaming vocab GEMM + online logsumexp + target/cluster gather.
// Block = 128 threads (4 waves), owns RT*16 rows x COLS_PER_BLOCK columns.
// A (RT*16 x d) staged in LDS (async DMA when available); each streamed
// B fragment feeds RT WMMAs.
// ---------------------------------------------------------------------------
template <int RT>
__global__ void vocab_pass(const float* __restrict__ A,   // proj [4096 x d]
                           const float* __restrict__ W,   // [Vmain x d] row-major
                           const float* __restrict__ Bv,  // bias [Vmain]
                           const float* __restrict__ CW,  // cluster weight [3 x d] or null
                           const float* __restrict__ CBv, // cluster bias [3] or null
                           const int*   __restrict__ tgt,
                           int d, int Vmain, int Vtot, int lidx, int ridx,
                           float* __restrict__ pmax, float* __restrict__ psum, int NB,
                           float* __restrict__ tlog,      // [4096] (this cluster)
                           float* __restrict__ clog)      // [4096 x 3] or null
{
  constexpr int BR = RT * 16;               // rows per block
  extern __shared__ float smem[];
  float* As   = smem;                       // BR*d floats
  int*   tgts = (int*)(smem + BR * d);      // BR ints
  float* wm   = smem + BR * d + BR;         // 4*BR
  float* wsv  = wm + 4 * BR;                // 4*BR

  const int tid     = threadIdx.x;
  const int rowbase = blockIdx.y * BR;

#if USE_ASYNC_LDS
  {
    // Direct memory->LDS DMA (GLOBAL_LOAD_ASYNC_TO_LDS_B128, ASYNCcnt-tracked):
    // BR consecutive proj rows are one contiguous 16B-aligned span.
    as1_v4i* gsrc = (as1_v4i*)(A + (size_t)rowbase * d);
    as3_v4i* ldst = (as3_v4i*)As;
    const int nvec = (BR * d) / 4;                    // 16-byte packets
    for (int idx = tid; idx < nvec; idx += blockDim.x)
      __builtin_amdgcn_global_load_async_to_lds_b128(gsrc + idx, ldst + idx, 0, 0);
#if __has_builtin(__builtin_amdgcn_s_wait_asynccnt)
    __builtin_amdgcn_s_wait_asynccnt(0);
#else
    asm volatile("s_wait_asynccnt 0" ::: "memory");
#endif
  }
#else
  for (int idx = tid; idx < BR * d; idx += blockDim.x)
    As[idx] = A[(size_t)rowbase * d + idx];           // BR rows contiguous
#endif
  if (tid < BR) tgts[tid] = tgt[rowbase + tid];
  __syncthreads();

  const int lane = tid & 31;
  const int wave = tid >> 5;
  const int M    = lane & 15;
  const int h    = lane >> 4;

  float rmx[RT][8], rsm[RT][8];
#pragma unroll
  for (int r = 0; r < RT; ++r)
#pragma unroll
    for (int j = 0; j < 8; ++j) { rmx[r][j] = NEG_BIG; rsm[r][j] = 0.f; }

  const int blockcol = blockIdx.x * COLS_PER_BLOCK;

  for (int ch = wave; ch < CHUNKS_PER_BLOCK; ch += 4) {
    const int cbase = blockcol + ch * 16;
    if (cbase >= Vtot) break;
    const int  c     = cbase + M;
    const bool valid = (c < Vtot);
    const float* wrow;
    float bias;
    if (c < Vmain)  { wrow = W + (size_t)c * d;            bias = Bv[c]; }
    else if (valid) { wrow = CW + (size_t)(c - Vmain) * d; bias = CBv[c - Vmain]; }
    else            { wrow = W;                            bias = 0.f; }

    // Prefetch this lane's next chunk's weight row (speculative, toward L2).
    const int cnxt = c + 64;
    if (cnxt < Vmain) __builtin_prefetch(W + (size_t)cnxt * d, 0, 1);

    const float* arow = As + M * d + 2 * h;   // row-tile 0; tiles stride 16*d
    const float* brow = wrow + 2 * h;

    v8f acc[RT];
#pragma unroll
    for (int r = 0; r < RT; ++r) acc[r] = (v8f){};

#pragma unroll 4
    for (int k0 = 0; k0 < d; k0 += 4) {
      const v2f b = *(const v2f*)(brow + k0);      // one B fragment ...
#pragma unroll
      for (int r = 0; r < RT; ++r) {               // ... feeds RT WMMAs
        const v2f a = *(const v2f*)(arow + (size_t)r * 16 * d + k0);
        acc[r] = __builtin_amdgcn_wmma_f32_16x16x4_f32(false, a, false, b,
                                                       (short)0, acc[r],
                                                       false, false);
      }
    }

#pragma unroll
    for (int r = 0; r < RT; ++r) {
#pragma unroll
      for (int j = 0; j < 8; ++j) {
        const float val = valid ? (acc[r][j] + bias) : NEG_BIG;
        const float m   = redmax16(val);
        const float e   = valid ? __expf(val - m) : 0.f;
        const float s   = redsum16(e);
        if (m > rmx[r][j]) { rsm[r][j] = rsm[r][j] * __expf(rmx[r][j] - m) + s; rmx[r][j] = m; }
        else               { rsm[r][j] += s * __expf(m - rmx[r][j]); }

        const int rit = r * 16 + j + 8 * h;        // row within block
        const int t   = tgts[rit];
        if (valid && t >= lidx && t < ridx && (t - lidx) == c)
          tlog[rowbase + rit] = val;               // unique owning lane
        if (clog && valid && c >= Vmain)
          clog[(size_t)(rowbase + rit) * 3 + (c - Vmain)] = val;
      }
    }
  }

  if ((lane & 15) == 0) {                          // value replicated in 16-lane group
#pragma unroll
    for (int r = 0; r < RT; ++r)
#pragma unroll
      for (int j = 0; j < 8; ++j) {
        const int rit = r * 16 + j + 8 * h;
        wm[wave * BR + rit]  = rmx[r][j];
        wsv[wave * BR + rit] = rsm[r][j];
      }
  }
  __syncthreads();
  if (tid < BR) {
    float m = NEG_BIG, s = 0.f;
    for (int w = 0; w < 4; ++w) {
      const float mw = wm[w * BR + tid], sw = wsv[w * BR + tid];
      if (sw > 0.f) {
        if (mw > m) { s = s * __expf(m - mw) + sw; m = mw; }
        else        { s += sw * __expf(mw - m); }
      }
    }
    pmax[(size_t)(rowbase + tid) * NB + blockIdx.x] = m;
    psum[(size_t)(rowbase + tid) * NB + blockIdx.x] = s;
  }
}

// ---------------------------------------------------------------------------
// Pass 3: combine per-slab partials -> lse, assemble NLL per token.
// ---------------------------------------------------------------------------
__global__ void finalize_nll(const int* __restrict__ tgt,
                             const float* __restrict__ pm0, const float* __restrict__ ps0, int NB0,
                             const float* __restrict__ pm1, const float* __restrict__ ps1, int NB1,
                             const float* __restrict__ pm2, const float* __restrict__ ps2, int NB2,
                             const float* __restrict__ pm3, const float* __restrict__ ps3, int NB3,
                             const float* __restrict__ tlog,   // [4][4096]
                             const float* __restrict__ clog,   // [4096][3]
                             float* __restrict__ out)
{
  const int r = blockIdx.x * blockDim.x + threadIdx.x;
  if (r >= N_ROWS) return;
  const float* pms[4] = {pm0, pm1, pm2, pm3};
  const float* pss[4] = {ps0, ps1, ps2, ps3};
  const int    nbs[4] = {NB0, NB1, NB2, NB3};
  float lse[4];
#pragma unroll
  for (int i = 0; i < 4; ++i) {
    float m = NEG_BIG, s = 0.f;
    for (int p = 0; p < nbs[i]; ++p) {
      const float mp = pms[i][(size_t)r * nbs[i] + p];
      const float sp = pss[i][(size_t)r * nbs[i] + p];
      if (sp > 0.f) {
        if (mp > m) { s = s * expf(m - mp) + sp; m = mp; }
        else        { s += sp * expf(mp - m); }
      }
    }
    lse[i] = m + logf(s);
  }
  const int t = tgt[r];
  float nll;
  if (t < 50000) {
    nll = lse[0] - tlog[r];
  } else {
    const int k = (t < 100000) ? 1 : ((t < 180000) ? 2 : 3);
    const float headlp = clog[(size_t)r * 3 + (3 - k)] - lse[0];  // head col "-k"
    const float taillp = tlog[(size_t)k * N_ROWS + r] - lse[k];
    nll = -(headlp + taillp);
  }
  out[r] = nll;
}

// ---------------------------------------------------------------------------
extern "C" void kernel_launch(void* const* d_in, const int* in_sizes, int n_in,
                              void* d_out, int out_size, void* d_ws, size_t ws_size,
                              hipStream_t stream) {
  (void)in_sizes; (void)n_in; (void)out_size; (void)ws_size;
  const float* hidden = (const float*)d_in[0];
  const int*   target = (const int*)d_in[1];
  const float* cw     = (const float*)d_in[2];
  const float* cb     = (const float*)d_in[3];
  const float* w0 = (const float*)d_in[4];
  const float* b0 = (const float*)d_in[5];
  const float* p0 = (const float*)d_in[6];
  const float* w1 = (const float*)d_in[7];
  const float* b1 = (const float*)d_in[8];
  const float* p1 = (const float*)d_in[9];
  const float* w2 = (const float*)d_in[10];
  const float* b2 = (const float*)d_in[11];
  const float* p2 = (const float*)d_in[12];
  const float* w3 = (const float*)d_in[13];
  const float* b3 = (const float*)d_in[14];
  const float* p3 = (const float*)d_in[15];

  const int V0 = 50003, V1 = 50000, V2 = 80000, V3 = 70000;
  const int NB0 = (V0 + COLS_PER_BLOCK - 1) / COLS_PER_BLOCK;   // 13
  const int NB1 = (V1 + COLS_PER_BLOCK - 1) / COLS_PER_BLOCK;   // 13
  const int NB2 = (V2 + COLS_PER_BLOCK - 1) / COLS_PER_BLOCK;   // 20
  const int NB3 = (V3 + COLS_PER_BLOCK - 1) / COLS_PER_BLOCK;   // 18

  float* ws = (float*)d_ws;
  size_t off = 0;
  float* proj0 = ws + off; off += (size_t)N_ROWS * 1024;
  float* proj1 = ws + off; off += (size_t)N_ROWS * 256;
  float* proj2 = ws + off; off += (size_t)N_ROWS * 64;
  float* proj3 = ws + off; off += (size_t)N_ROWS * 16;
  float* pm0 = ws + off; off += (size_t)N_ROWS * NB0;
  float* ps0 = ws + off; off += (size_t)N_ROWS * NB0;
  float* pm1 = ws + off; off += (size_t)N_ROWS * NB1;
  float* ps1 = ws + off; off += (size_t)N_ROWS * NB1;
  float* pm2 = ws + off; off += (size_t)N_ROWS * NB2;
  float* ps2 = ws + off; off += (size_t)N_ROWS * NB2;
  float* pm3 = ws + off; off += (size_t)N_ROWS * NB3;
  float* ps3 = ws + off; off += (size_t)N_ROWS * NB3;
  float* tlog = ws + off; off += (size_t)4 * N_ROWS;
  float* clog = ws + off; off += (size_t)N_ROWS * 3;

  // Pass 1: projections (fp32 WMMA).
  proj_gemm<<<dim3(256 * 64), 32, 0, stream>>>(hidden, p0, proj0, 1024, 1024);
  proj_gemm<<<dim3(256 * 16), 32, 0, stream>>>(hidden, p1, proj1, 1024, 256);
  proj_gemm<<<dim3(256 * 4),  32, 0, stream>>>(hidden, p2, proj2, 1024, 64);
  proj_gemm<<<dim3(256 * 1),  32, 0, stream>>>(hidden, p3, proj3, 1024, 16);

  // Dynamic LDS: BR*d floats + BR targets + 8*BR combine scratch.
  auto shbytes = [](int RT, int d) {
    const int BR = RT * 16;
    return (size_t)(BR * d + BR + 8 * BR) * 4;
  };
  (void)hipFuncSetAttribute((const void*)vocab_pass<2>,
                            hipFuncAttributeMaxDynamicSharedMemorySize,
                            (int)shbytes(2, 1024));   // ~129 KB (2 WGs / 320KB WGP)
  (void)hipFuncSetAttribute((const void*)vocab_pass<4>,
                            hipFuncAttributeMaxDynamicSharedMemorySize,
                            (int)shbytes(4, 256));    // ~66 KB

  // Pass 2: streaming vocab logsumexp per cluster.
  vocab_pass<2><<<dim3(NB0, N_ROWS / 32), 128, shbytes(2, 1024), stream>>>(
      proj0, w0, b0, cw, cb, target, 1024, 50000, V0, 0, 50000,
      pm0, ps0, NB0, tlog + 0 * N_ROWS, clog);
  vocab_pass<4><<<dim3(NB1, N_ROWS / 64), 128, shbytes(4, 256), stream>>>(
      proj1, w1, b1, nullptr, nullptr, target, 256, V1, V1, 50000, 100000,
      pm1, ps1, NB1, tlog + 1 * N_ROWS, nullptr);
  vocab_pass<4><<<dim3(NB2, N_ROWS / 64), 128, shbytes(4, 64), stream>>>(
      proj2, w2, b2, nullptr, nullptr, target, 64, V2, V2, 100000, 180000,
      pm2, ps2, NB2, tlog + 2 * N_ROWS, nullptr);
  vocab_pass<4><<<dim3(NB3, N_ROWS / 64), 128, shbytes(4, 16), stream>>>(
      proj3, w3, b3, nullptr, nullptr, target, 16, V3, V3, 180000, 250000,
      pm3, ps3, NB3, tlog + 3 * N_ROWS, nullptr);

  // Pass 3: finalize NLL.
  finalize_nll<<<dim3(N_ROWS / 256), 256, 0, stream>>>(
      target, pm0, ps0, NB0, pm1, ps1, NB1, pm2, ps2, NB2, pm3, ps3, NB3,
      tlog, clog, (float*)d_out);
}